// TextEncoder_14422500180139
// MI455X (gfx1250) — compile-verified
//
#include <hip/hip_runtime.h>

// ---------------------------------------------------------------------------
// Fused embedding + x@W + LSTM recurrence persistent kernel for gfx1250.
// FP32 WMMA (V_WMMA_F32_16X16X4_F32); W/U column slices staged once in LDS
// (192 KB of the 320 KB WGP LDS), h state ping-pongs through L2.
// ---------------------------------------------------------------------------

typedef __attribute__((ext_vector_type(2))) float v2f;
typedef __attribute__((ext_vector_type(8))) float v8f;

#define VOCAB  50257
#define EMBED  256
#define HIDDEN 512
#define BATCH  64
#define SEQ    512
#define NG4    (4 * HIDDEN)          // 2048 gate columns

#define NWG         32               // persistent workgroups
#define COLS_PER_WG (HIDDEN / NWG)   // 16 h-columns per WG
#define BLOCK       128              // 4 wave32 waves
#define LCOLS       64               // 4 gates * 16 local columns
#define LDS_FLOATS  ((EMBED + HIDDEN) * LCOLS)   // 49152 floats = 192 KB

__global__ void lstm_init_kernel(unsigned* bar) { bar[0] = 0u; }

__device__ __forceinline__ float sigf(float x) {
  return 1.0f / (1.0f + __expf(-x));
}

__device__ __forceinline__ float fast_tanh(float x) {
#if __has_builtin(__builtin_amdgcn_tanhf)
  return __builtin_amdgcn_tanhf(x);     // v_tanh_f32 (gfx1250 TRANS unit)
#else
  return tanhf(x);
#endif
}

__device__ __forceinline__ void grid_barrier(unsigned* bar, unsigned target) {
  __threadfence();                 // make this thread's h stores device-visible
  __syncthreads();
  if (threadIdx.x == 0) {
    __hip_atomic_fetch_add(bar, 1u, __ATOMIC_RELEASE, __HIP_MEMORY_SCOPE_AGENT);
    while (__hip_atomic_load(bar, __ATOMIC_ACQUIRE, __HIP_MEMORY_SCOPE_AGENT) < target) {
      __builtin_amdgcn_s_sleep(1);
    }
  }
  __syncthreads();
  __threadfence();                 // acquire: invalidate stale near caches
}

__global__ __launch_bounds__(BLOCK) void lstm_persistent_kernel(
    const int*   __restrict__ tok,    // [BATCH][SEQ]
    const float* __restrict__ embed,  // [VOCAB][EMBED]
    const float* __restrict__ W,      // [EMBED][NG4]
    const float* __restrict__ U,      // [HIDDEN][NG4]
    const float* __restrict__ bias,   // [NG4]
    float*       __restrict__ out,    // [BATCH][HIDDEN]
    float*       __restrict__ hbuf,   // [2][HIDDEN/2][BATCH][2]  (k-pair layout)
    unsigned*    __restrict__ bar)
{
  extern __shared__ float smem[];
  float* lw = smem;                  // [EMBED/2][LCOLS][2]  =  64 KB
  float* lu = smem + EMBED * LCOLS;  // [HIDDEN/2][LCOLS][2] = 128 KB

  const int wg   = blockIdx.x;
  const int tid  = threadIdx.x;
  const int lane = tid & 31;
  const int wave = tid >> 5;            // 0..3  -> batch row tile
  const int m0   = wave * 16;
  const int hi   = lane >> 4;           // 0 / 1 (lane half)
  const int l15  = lane & 15;
  const int mrow = m0 + l15;            // A-fragment row owned by this lane
  const int n0   = wg * COLS_PER_WG;    // h-column slice base of this WG
  const int ncol = n0 + l15;            // C/D fragment column of this lane

  float* h0 = hbuf;
  float* h1 = hbuf + HIDDEN * BATCH;

  // ---- stage W and U column slices into LDS (pair-interleaved: one
  //      ds_load_b64 yields the {k, k+1} B-fragment this lane needs) ----
  for (int idx = tid; idx < EMBED * LCOLS; idx += BLOCK) {
    const int k   = idx >> 6;                                  // 0..255
    const int lc  = idx & (LCOLS - 1);                         // local col
    const int col = (lc >> 4) * HIDDEN + n0 + (lc & 15);       // gate col
    lw[(k >> 1) * (2 * LCOLS) + lc * 2 + (k & 1)] = W[(size_t)k * NG4 + col];
  }
  for (int idx = tid; idx < HIDDEN * LCOLS; idx += BLOCK) {
    const int k   = idx >> 6;                                  // 0..511
    const int lc  = idx & (LCOLS - 1);
    const int col = (lc >> 4) * HIDDEN + n0 + (lc & 15);
    lu[(k >> 1) * (2 * LCOLS) + lc * 2 + (k & 1)] = U[(size_t)k * NG4 + col];
  }

  // cooperatively zero initial h (buffer 0)
  for (int i = wg * BLOCK + tid; i < HIDDEN * BATCH; i += NWG * BLOCK)
    h0[i] = 0.0f;

  const v8f vzero = {0.f, 0.f, 0.f, 0.f, 0.f, 0.f, 0.f, 0.f};
  v8f creg = vzero;                      // persistent cell state tile
  v8f hreg = vzero;                      // persistent hidden state tile

  float bb[4];
#pragma unroll
  for (int gi = 0; gi < 4; ++gi) bb[gi] = bias[gi * HIDDEN + ncol];

  grid_barrier(bar, NWG);                // h0 zeroed + LDS staged everywhere

  for (int t = 0; t < SEQ; ++t) {
    const float* cur = (t & 1) ? h1 : h0;
    float*       nxt = (t & 1) ? h0 : h1;

    v8f acc[4];
#pragma unroll
    for (int gi = 0; gi < 4; ++gi) acc[gi] = vzero;

    // ---- x_t @ W : A gathered from embedding row (v2f = {k, k+1}) ----
    const v2f* erow2 =
        (const v2f*)(embed + (size_t)tok[mrow * SEQ + t] * EMBED);
#pragma unroll 4
    for (int kk = 0; kk < EMBED; kk += 4) {
      const int kp = (kk >> 1) + hi;         // pair index (kb = 2*kp even)
      const v2f av = erow2[kp];
#pragma unroll
      for (int gi = 0; gi < 4; ++gi) {
        const v2f bv =
            *(const v2f*)&lw[kp * (2 * LCOLS) + (gi * 16 + l15) * 2];
        acc[gi] = __builtin_amdgcn_wmma_f32_16x16x4_f32(
            false, av, false, bv, (short)0, acc[gi], false, false);
      }
    }

    // ---- h_{t-1} @ U : A from k-pair ping-pong buffer (global/L2) ----
    const v2f* curp = (const v2f*)cur;       // [k/2][BATCH] of float2
#pragma unroll 4
    for (int kk = 0; kk < HIDDEN; kk += 4) {
      const int kp = (kk >> 1) + hi;
      const v2f av = curp[kp * BATCH + mrow];
#pragma unroll
      for (int gi = 0; gi < 4; ++gi) {
        const v2f bv =
            *(const v2f*)&lu[kp * (2 * LCOLS) + (gi * 16 + l15) * 2];
        acc[gi] = __builtin_amdgcn_wmma_f32_16x16x4_f32(
            false, av, false, bv, (short)0, acc[gi], false, false);
      }
    }

    // prefetch next step's embedding row into L2 while gates compute
    if (t + 1 < SEQ)
      __builtin_prefetch(embed + (size_t)tok[mrow * SEQ + t + 1] * EMBED, 0, 0);

    // ---- gates + masked state update (i,f,g,o tiles are wave-local) ----
#pragma unroll
    for (int v = 0; v < 8; ++v) {
      const int row = m0 + v + hi * 8;          // C/D layout: M = vgpr + 8*half
      const bool msk = tok[row * SEQ + t] != 0;
      const float zi = acc[0][v] + bb[0];
      const float zf = acc[1][v] + bb[1];
      const float zg = acc[2][v] + bb[2];
      const float zo = acc[3][v] + bb[3];
      const float ig = sigf(zi);
      const float fg = sigf(zf);
      const float gg = fast_tanh(zg);
      const float og = sigf(zo);
      const float cn = fg * creg[v] + ig * gg;
      const float hn = og * fast_tanh(cn);
      if (msk) { creg[v] = cn; hreg[v] = hn; }
      // publish masked h in k-pair layout: element (k=ncol, b=row)
      nxt[(ncol >> 1) * (2 * BATCH) + row * 2 + (ncol & 1)] = hreg[v];
    }

    grid_barrier(bar, (unsigned)(NWG * (t + 2)));
  }

  // final h -> output (each (wg, wave, lane, v) owns a unique element)
#pragma unroll
  for (int v = 0; v < 8; ++v) {
    const int row = m0 + v + hi * 8;
    out[row * HIDDEN + ncol] = hreg[v];
  }
}

extern "C" void kernel_launch(void* const* d_in, const int* in_sizes, int n_in,
                              void* d_out, int out_size, void* d_ws, size_t ws_size,
                              hipStream_t stream) {
  const int*   tok   = (const int*)d_in[0];
  const float* embed = (const float*)d_in[1];
  const float* W     = (const float*)d_in[2];
  const float* U     = (const float*)d_in[3];
  const float* bias  = (const float*)d_in[4];
  float*       out   = (float*)d_out;

  float*    hbuf = (float*)d_ws;                                   // 256 KB
  unsigned* bar  = (unsigned*)((char*)d_ws +
                               (size_t)2 * HIDDEN * BATCH * sizeof(float));

  const size_t lds_bytes = (size_t)LDS_FLOATS * sizeof(float);     // 192 KB
  (void)hipFuncSetAttribute((const void*)lstm_persistent_kernel,
                            hipFuncAttributeMaxDynamicSharedMemorySize,
                            (int)lds_bytes);

  lstm_init_kernel<<<1, 1, 0, stream>>>(bar);
  lstm_persistent_kernel<<<NWG, BLOCK, lds_bytes, stream>>>(
      tok, embed, W, U, bias, out, hbuf, bar);
}